// MultiheadAttention_23124103921928
// MI455X (gfx1250) — compile-verified
//
#include <hip/hip_runtime.h>
#include <math.h>

// CDNA5 / gfx1250, wave32. All matrix math via V_WMMA_F32_16X16X4_F32 (f32-exact).
typedef float v2f __attribute__((ext_vector_type(2)));
typedef float v8f __attribute__((ext_vector_type(8)));

#define FMIN_F32 (-3.4028234663852886e38f)   // torch.finfo(float32).min (nan_to_num of -inf)

// ---------------------------------------------------------------------------
// Kernel 1: row sums.  qs[b,i] = 0.25 * sum_f query[b,i,f]  (1/scale folded in)
//           ks[b,j]   = sum_f key[b,j,f]
//           vs[b,j]   = sum_f value[b,j,f]
// 12288 rows total (3 tensors * 4 batches * 1024 rows), one block per row.
// ---------------------------------------------------------------------------
__global__ __launch_bounds__(256) void row_reduce_kernel(
    const float* __restrict__ q, const float* __restrict__ k,
    const float* __restrict__ v, float* __restrict__ red)
{
    const int row   = blockIdx.x;        // 0 .. 12287
    const int which = row >> 12;         // 0:q 1:k 2:v
    const int r     = row & 4095;        // b*1024 + i
    const float* src = (which == 0) ? q : (which == 1) ? k : v;
    const float* p   = src + (size_t)r * 1024;

    float s = 0.f;
    for (int c = threadIdx.x; c < 1024; c += 256) s += p[c];

    __shared__ float smem[256];
    smem[threadIdx.x] = s;
    __syncthreads();
    for (int off = 128; off > 0; off >>= 1) {
        if (threadIdx.x < off) smem[threadIdx.x] += smem[threadIdx.x + off];
        __syncthreads();
    }
    if (threadIdx.x == 0) {
        float val = smem[0];
        if (which == 0) val *= 0.25f;    // scores / sqrt(H)=4 folded into qsum
        red[which * 4096 + r] = val;
    }
}

// ---------------------------------------------------------------------------
// Kernel 2: fused attention per (head h, 16-row i-tile).  One wave per block.
//   loop over 64 j-tiles:
//     G(16x16)  = q_w[h,i-tile,:] @ k_w[h,j-tile,:]^T       (WMMA f32, K=64)
//     scores[b] = qs[b,i]*ks[b,j]*G ; mask(causal | pad[b,j]) -> FLT_MIN
//     softmax over b (4 values, in registers)
//     A[b]      = weight[b]*vs[b,j]  -> LDS (C-layout -> A-layout transit)
//     acc[b,t]  += A[b](16x16) @ v_w[h,j-tile, t-chunk]      (WMMA f32)
//   store heads directly in the reshaped layout flat[b, h*64+it, m*64+t].
// ---------------------------------------------------------------------------
__global__ __launch_bounds__(32) void attn_fused_kernel(
    const float* __restrict__ qw, const float* __restrict__ kw,
    const float* __restrict__ vw, const int* __restrict__ pad,
    const float* __restrict__ red, float* __restrict__ flat)
{
    const int it   = blockIdx.x;     // i-tile 0..63
    const int h    = blockIdx.y;     // head 0..15
    const int lane = threadIdx.x;    // 0..31 (wave32)
    const int hi   = lane >> 4;      // lane half
    const int ln   = lane & 15;
    const int koff = hi * 2;         // A/B f32 K split: {0,1} | {2,3}

    const float* qs = red;
    const float* ks = red + 4096;
    const float* vs = red + 8192;

    const int i0 = it * 16;
    const float* qwh = qw + (size_t)h * 1024 * 64;
    const float* kwh = kw + (size_t)h * 1024 * 64;
    const float* vwh = vw + (size_t)h * 1024 * 64;

    // hoist qs for this lane's 8 output rows (C-layout: M = e + 8*hi)
    float qrow[4][8];
    for (int b = 0; b < 4; ++b)
        for (int e = 0; e < 8; ++e)
            qrow[b][e] = qs[b * 1024 + i0 + e + 8 * hi];

    __shared__ float tA[4][16][16];  // per-batch A tiles (weight * vsum)

    v8f acc[4][4];                   // [batch][t-chunk]  -> 128 VGPRs
    for (int b = 0; b < 4; ++b)
        for (int tc = 0; tc < 4; ++tc) acc[b][tc] = (v8f){};

    for (int jt = 0; jt < 64; ++jt) {
        const int j0 = jt * 16;

        // ---- G tile: 16x16, K=64 in 16 WMMA steps of K=4 -------------------
        v8f g = (v8f){};
        const float* arow = qwh + (size_t)(i0 + ln) * 64 + koff;
        const float* brow = kwh + (size_t)(j0 + ln) * 64 + koff;
        for (int kb = 0; kb < 64; kb += 4) {
            v2f a  = *(const v2f*)(arow + kb);   // A[ln][kb+koff..+1]
            v2f bf = *(const v2f*)(brow + kb);   // B[kb+koff..+1][ln] = k_w^T
            g = __builtin_amdgcn_wmma_f32_16x16x4_f32(
                    false, a, false, bf, (short)0, g, false, false);
        }

        // ---- batch softmax + mask, build A = weight * vsum into LDS --------
        const int jcol = j0 + ln;                 // this lane's column j
        float ksb[4], vsb[4];
        bool  pmb[4];
        for (int b = 0; b < 4; ++b) {
            ksb[b] = ks[b * 1024 + jcol];
            vsb[b] = vs[b * 1024 + jcol];
            pmb[b] = (pad[b * 1024 + jcol] == 1);
        }
        for (int e = 0; e < 8; ++e) {
            const int m = e + 8 * hi;
            const bool causal = (jcol > i0 + m);
            float s[4];
            float mx = FMIN_F32;
            for (int b = 0; b < 4; ++b) {
                float sc = (causal || pmb[b]) ? FMIN_F32
                                              : qrow[b][e] * ksb[b] * g[e];
                s[b] = sc;
                mx   = fmaxf(mx, sc);
            }
            float w[4], sum = 0.f;
            for (int b = 0; b < 4; ++b) { w[b] = __expf(s[b] - mx); sum += w[b]; }
            const float inv = 1.f / sum;
            for (int b = 0; b < 4; ++b)
                tA[b][m][ln] = w[b] * inv * vsb[b];   // fold vsum[b,j] into A
        }
        __syncthreads();

        // ---- PV: acc[b][tc] += A[b] @ v_w[h, j-tile, tc*16..]  (K=16) ------
        for (int b = 0; b < 4; ++b) {
            for (int kc = 0; kc < 4; ++kc) {
                v2f a = *(const v2f*)&tA[b][ln][kc * 4 + koff];
                const float* bp = vwh + (size_t)(j0 + kc * 4 + koff) * 64 + ln;
                for (int tc = 0; tc < 4; ++tc) {
                    v2f bf;
                    bf.x = bp[tc * 16];          // K = kc*4+koff,   N = tc*16+ln
                    bf.y = bp[tc * 16 + 64];     // K = kc*4+koff+1
                    acc[b][tc] = __builtin_amdgcn_wmma_f32_16x16x4_f32(
                        false, a, false, bf, (short)0, acc[b][tc], false, false);
                }
            }
        }
        __syncthreads();
    }

    // ---- store heads in reshaped layout: flat[b, h*64+it, m*64 + t] --------
    const int r = h * 64 + it;
    for (int b = 0; b < 4; ++b)
        for (int tc = 0; tc < 4; ++tc)
            for (int e = 0; e < 8; ++e) {
                const int m = e + 8 * hi;
                const int t = tc * 16 + ln;
                flat[((size_t)b * 1024 + r) * 1024 + m * 64 + t] = acc[b][tc][e];
            }
}

// ---------------------------------------------------------------------------
// Kernel 3: out[b] = flat[b] (1024x1024) @ o_w (1024x1024), f32 WMMA.
// One wave per 16x16 output tile, K-loop 1024 in steps of 4.
// ---------------------------------------------------------------------------
__global__ __launch_bounds__(32) void out_gemm_kernel(
    const float* __restrict__ flat, const float* __restrict__ ow,
    float* __restrict__ out)
{
    const int ot = blockIdx.x, rt = blockIdx.y, b = blockIdx.z;
    const int lane = threadIdx.x;
    const int hi = lane >> 4, ln = lane & 15, koff = hi * 2;

    const float* A  = flat + (size_t)b * 1024 * 1024
                           + (size_t)(rt * 16 + ln) * 1024 + koff;
    const float* Bm = ow + ot * 16 + ln;

    v8f c = (v8f){};
    for (int k = 0; k < 1024; k += 4) {
        if ((k & 63) == 0) {
            __builtin_prefetch(A + k + 64, 0, 0);                    // global_prefetch
            __builtin_prefetch(Bm + (size_t)(k + 64) * 1024, 0, 0);
        }
        v2f a = *(const v2f*)(A + k);
        v2f bf;
        bf.x = Bm[(size_t)(k + koff) * 1024];
        bf.y = Bm[(size_t)(k + koff + 1) * 1024];
        c = __builtin_amdgcn_wmma_f32_16x16x4_f32(
                false, a, false, bf, (short)0, c, false, false);
    }
    for (int e = 0; e < 8; ++e) {
        const int m = e + 8 * hi;
        out[((size_t)b * 1024 + rt * 16 + m) * 1024 + ot * 16 + ln] = c[e];
    }
}

// ---------------------------------------------------------------------------
extern "C" void kernel_launch(void* const* d_in, const int* in_sizes, int n_in,
                              void* d_out, int out_size, void* d_ws, size_t ws_size,
                              hipStream_t stream) {
    const float* query = (const float*)d_in[0];
    const float* key   = (const float*)d_in[1];
    const float* value = (const float*)d_in[2];
    const int*   pad   = (const int*)  d_in[3];
    const float* q_w   = (const float*)d_in[4];
    const float* k_w   = (const float*)d_in[5];
    const float* v_w   = (const float*)d_in[6];
    const float* o_w   = (const float*)d_in[7];
    float* out = (float*)d_out;

    float* red  = (float*)d_ws;          // 12288 f32: qs | ks | vs
    float* flat = red + 12288;           // 4M f32: reshaped heads

    row_reduce_kernel<<<dim3(12288), dim3(256), 0, stream>>>(query, key, value, red);
    attn_fused_kernel<<<dim3(64, 16), dim3(32), 0, stream>>>(q_w, k_w, v_w, pad, red, flat);
    out_gemm_kernel  <<<dim3(64, 64, 4), dim3(32), 0, stream>>>(flat, o_w, out);
}